// DeformConvWarp_74251394613305
// MI455X (gfx1250) — compile-verified
//
#include <hip/hip_runtime.h>
#include <hip/hip_bf16.h>

typedef __attribute__((ext_vector_type(2))) float v2f;
typedef __attribute__((ext_vector_type(8))) float v8f;

#define C_IN   128
#define CO     128
#define HW     96
#define PIX    (HW * HW)      // 9216 output pixels per image
#define K2     9
#define PTILE  64             // pixels per block (4 x 16-wide WMMA column tiles)
#define NTILE  4
#define TILE_WORDS 2080       // 32 ksteps * 32 lanes * 2 + 32 pad (pad => tiles offset 32 banks)
#define NIMG   8
#define PACK_ELEMS (8 * K2 * 32 * 64)   // oblocks * taps * ksteps * (32 lanes * 2 floats) = 147456

__device__ __forceinline__ int iclamp(int v, int lo, int hi) {
    return v < lo ? lo : (v > hi ? hi : v);
}

// Pre-pack weight [Co=128][C=128][3][3] into per-lane A-fragment order:
// packW[((ob*9 + t)*32 + ks)*64 + lane*2 + j] = W[o = ob*16 + (lane&15)][c = ks*4 + (lane>>4)*2 + j][tap t]
__global__ __launch_bounds__(256)
void dcn_pack_weight_kernel(const float* __restrict__ w, float* __restrict__ packW) {
    int idx = blockIdx.x * 256 + threadIdx.x;
    if (idx >= PACK_ELEMS) return;
    int j    = idx & 1;
    int lane = (idx >> 1) & 31;
    int ks   = (idx >> 6) & 31;
    int t    = (idx >> 11) % K2;
    int ob   = idx / (K2 * 2048);
    int o = ob * 16 + (lane & 15);
    int c = ks * 4 + ((lane >> 4) << 1) + j;
    packW[idx] = w[(o * C_IN + c) * K2 + t];
}

__global__ __launch_bounds__(256)
void dcn_main_kernel(const float* __restrict__ x,
                     const float* __restrict__ offset,
                     const float* __restrict__ mask,
                     const float* __restrict__ packW,
                     const float* __restrict__ bias,
                     float* __restrict__ out) {
    // B tile stored directly in WMMA B-fragment order:
    // Bf[tile*TILE_WORDS + ks*64 + lane*2 + j]  (lane = half*16 + col, j selects K pair element)
    __shared__ float Bf[NTILE * TILE_WORDS];      // 33.3 KB
    __shared__ int   sIdx[K2][PTILE][4];          // 9 KB   bilinear corner indices
    __shared__ float sWt[K2][PTILE][4];           // 9 KB   mask-premultiplied corner weights

    const int n       = blockIdx.y;
    const int pixBase = blockIdx.x * PTILE;
    const int tid     = threadIdx.x;
    const int wave    = tid >> 5;                 // 0..7 -> output-row strip (16 rows)
    const int lane    = tid & 31;
    const int col     = lane & 15;                // N index within a 16-wide tile

    // ---- Phase 0: bilinear sampling parameters for all (tap, pixel) ----
    for (int it = tid; it < K2 * PTILE; it += 256) {
        int p  = it & (PTILE - 1);
        int t  = it >> 6;
        int gp = pixBase + p;
        int oy = gp / HW, ox = gp % HW;
        float offy = offset[((size_t)(n * 2 * K2 + 2 * t    )) * PIX + gp];
        float offx = offset[((size_t)(n * 2 * K2 + 2 * t + 1)) * PIX + gp];
        float mm   = mask  [((size_t)(n * K2 + t)) * PIX + gp];
        int ky = t / 3, kx = t % 3;
        float py = (float)(oy - 1 + ky) + offy;   // STRIDE=1, PAD=1, DIL=1
        float px = (float)(ox - 1 + kx) + offx;
        float y0f = floorf(py), x0f = floorf(px);
        float fy = py - y0f,    fx = px - x0f;
        int iy0 = (int)y0f, ix0 = (int)x0f;
        int iy1 = iy0 + 1,  ix1 = ix0 + 1;
        bool vy0 = (iy0 >= 0) & (iy0 < HW);
        bool vy1 = (iy1 >= 0) & (iy1 < HW);
        bool vx0 = (ix0 >= 0) & (ix0 < HW);
        bool vx1 = (ix1 >= 0) & (ix1 < HW);
        int cy0 = iclamp(iy0, 0, HW - 1), cy1 = iclamp(iy1, 0, HW - 1);
        int cx0 = iclamp(ix0, 0, HW - 1), cx1 = iclamp(ix1, 0, HW - 1);
        sIdx[t][p][0] = cy0 * HW + cx0;
        sIdx[t][p][1] = cy0 * HW + cx1;
        sIdx[t][p][2] = cy1 * HW + cx0;
        sIdx[t][p][3] = cy1 * HW + cx1;
        sWt[t][p][0] = (vy0 && vx0) ? (1.f - fy) * (1.f - fx) * mm : 0.f;
        sWt[t][p][1] = (vy0 && vx1) ? (1.f - fy) * fx * mm        : 0.f;
        sWt[t][p][2] = (vy1 && vx0) ? fy * (1.f - fx) * mm        : 0.f;
        sWt[t][p][3] = (vy1 && vx1) ? fy * fx * mm                : 0.f;
    }

    v8f acc0 = {}, acc1 = {}, acc2 = {}, acc3 = {};

    // Gather-phase constants: p and dlt are fixed per thread across the whole kernel.
    const int gp_  = tid & (PTILE - 1);           // pixel within tile
    const int c0   = tid >> 6;                    // 0..3: channel phase (== c & 3 for all iters)
    // LDS fragment write slot for (c = c0 + 4*i, p):
    //   tile = p>>4, half = c0>>1, j = c0&1, word = tile*TILE_WORDS + i*64 + (half*16 + (p&15))*2 + j
    const int wrBase = (gp_ >> 4) * TILE_WORDS + (((c0 >> 1) << 4) + (gp_ & 15)) * 2 + (c0 & 1);

    for (int t = 0; t < K2; ++t) {
        __syncthreads();   // protect Bf reuse from previous tap's GEMM
        {
            int i0 = sIdx[t][gp_][0], i1 = sIdx[t][gp_][1], i2 = sIdx[t][gp_][2], i3 = sIdx[t][gp_][3];
            float w0 = sWt[t][gp_][0], w1 = sWt[t][gp_][1], w2 = sWt[t][gp_][2], w3 = sWt[t][gp_][3];
            const float* xp = x + ((size_t)(n * C_IN + c0)) * PIX;
            #pragma unroll 4
            for (int i = 0; i < 32; ++i) {        // channel c = c0 + 4*i
                float v = xp[i0] * w0 + xp[i1] * w1 + xp[i2] * w2 + xp[i3] * w3;
                Bf[wrBase + i * 64] = v;
                xp += 4 * PIX;
            }
        }
        __syncthreads();

        const float* pA = packW + ((size_t)(wave * K2 + t)) * 32 * 64;
        const float* pB = Bf + lane * 2;
        for (int ks = 0; ks < 32; ++ks) {
            v2f a = *(const v2f*)(pA + ks * 64 + lane * 2);
            v2f b0 = *(const v2f*)(pB + ks * 64);
            v2f b1 = *(const v2f*)(pB + ks * 64 + TILE_WORDS);
            v2f b2 = *(const v2f*)(pB + ks * 64 + 2 * TILE_WORDS);
            v2f b3 = *(const v2f*)(pB + ks * 64 + 3 * TILE_WORDS);
            acc0 = __builtin_amdgcn_wmma_f32_16x16x4_f32(false, a, false, b0, (short)0, acc0, false, false);
            acc1 = __builtin_amdgcn_wmma_f32_16x16x4_f32(false, a, false, b1, (short)0, acc1, false, false);
            acc2 = __builtin_amdgcn_wmma_f32_16x16x4_f32(false, a, false, b2, (short)0, acc2, false, false);
            acc3 = __builtin_amdgcn_wmma_f32_16x16x4_f32(false, a, false, b3, (short)0, acc3, false, false);
        }
    }

    // ---- Epilogue: C/D layout VGPR r -> row (r | r+8), N = lane&15 ----
    const int oBase = wave * 16 + ((lane >> 4) << 3);
    for (int r = 0; r < 8; ++r) {
        int o = oBase + r;
        float bv = bias[o];
        size_t base = ((size_t)(n * CO + o)) * PIX + pixBase;
        out[base +      col] = acc0[r] + bv;
        out[base + 16 + col] = acc1[r] + bv;
        out[base + 32 + col] = acc2[r] + bv;
        out[base + 48 + col] = acc3[r] + bv;
    }
}

extern "C" void kernel_launch(void* const* d_in, const int* in_sizes, int n_in,
                              void* d_out, int out_size, void* d_ws, size_t ws_size,
                              hipStream_t stream) {
    const float* x      = (const float*)d_in[0];   // [8,128,96,96]
    const float* offset = (const float*)d_in[1];   // [8,18,96,96]
    const float* mask   = (const float*)d_in[2];   // [8,9,96,96]
    const float* weight = (const float*)d_in[3];   // [128,128,3,3]
    const float* bias   = (const float*)d_in[4];   // [128]
    float* out   = (float*)d_out;                  // [8,128,96,96]
    float* packW = (float*)d_ws;                   // needs 576 KB scratch

    dcn_pack_weight_kernel<<<(PACK_ELEMS + 255) / 256, 256, 0, stream>>>(weight, packW);

    dim3 grid(PIX / PTILE, NIMG);                  // 144 x 8 blocks
    dcn_main_kernel<<<grid, 256, 0, stream>>>(x, offset, mask, packW, bias, out);
}